// BahdanauAttention_37864431682259
// MI455X (gfx1250) — compile-verified
//
#include <hip/hip_runtime.h>
#include <math.h>

// Problem sizes (fixed by the reference)
#define B_ 32
#define S_ 4096
#define D_ 512
#define U_ 512

// Row-blocking for the WMMA kernel: 4 row-tiles of 16 => 64 rows per WG
#define RT_ 4
#define ROWS_ (RT_ * 16)

typedef __attribute__((ext_vector_type(2))) float v2f;
typedef __attribute__((ext_vector_type(8))) float v8f;

// ---------------------------------------------------------------------------
// K0: q[b,u] = dec[b,:] @ W2[:,u] + b1[u] + b2[u]   (both biases folded once)
// ---------------------------------------------------------------------------
__global__ __launch_bounds__(256) void dec_proj_kernel(
    const float* __restrict__ dec, const float* __restrict__ W2,
    const float* __restrict__ b1, const float* __restrict__ b2,
    float* __restrict__ qbuf)
{
    const int b = blockIdx.x >> 1;
    const int u = ((blockIdx.x & 1) << 8) + threadIdx.x;
    float acc = b1[u] + b2[u];
    const float* drow = dec + b * D_;
#pragma unroll 8
    for (int d = 0; d < D_; ++d)
        acc += drow[d] * W2[(size_t)d * U_ + u];
    qbuf[b * U_ + u] = acc;
}

// ---------------------------------------------------------------------------
// K1: fused  score[b,s] = V . tanh(enc[b,s,:] @ W1 + q[b,:]) + bv
// One workgroup (8 waves, wave32) per 64-row slab (4 row-tiles of 16).
// A slab (64x512 f32, 128KB) staged in LDS. Each wave owns 4 of the 32
// u-tiles; per k-step it loads 4 A-fragments (ds_load_b64) + 4x2 B-fragments
// (global b32, L2-resident W1) and issues 16 V_WMMA_F32_16X16X4_F32 —
// each B fragment is reused across the 4 row-tiles (4x less L2 traffic).
// Epilogue fuses tanh + dot(V), shfl_xor reduce over N, LDS combine.
// ---------------------------------------------------------------------------
__global__ __launch_bounds__(256) void attn_score_kernel(
    const float* __restrict__ enc,   // [B,S,D]
    const float* __restrict__ W1,    // [D,U]
    const float* __restrict__ qbuf,  // [B,U]
    const float* __restrict__ V,     // [U]
    const float* __restrict__ bv,    // [1]
    float* __restrict__ scores)      // [B,S]
{
    __shared__ float As[ROWS_ * D_];   // 128 KB (CDNA5: up to 320KB/WG)
    __shared__ float sc[ROWS_];

    const int tid = threadIdx.x;
    const long rowBase = (long)blockIdx.x * ROWS_;  // global BS row (b*S + s0)
    const int b = (int)(rowBase / S_);              // S % 64 == 0 -> single b

    // Stage the 64x512 A slab, float4-vectorized (256 thr x 32 x b128).
    const float4* src4 = (const float4*)(enc + rowBase * D_);
    float4* As4 = (float4*)As;
#pragma unroll
    for (int i = 0; i < (ROWS_ * D_ / 4) / 256; ++i)
        As4[tid + i * 256] = src4[tid + i * 256];
    if (tid < ROWS_) sc[tid] = 0.0f;
    __syncthreads();

    const int wave = tid >> 5;
    const int lane = tid & 31;
    const int half = lane >> 4;     // K-pair selector (ISA 7.12.2 layout)
    const int l16  = lane & 15;     // M for A-frag, N for B/C-frag

    // 4 u-tiles per wave: tiles {wave, wave+8, wave+16, wave+24}
    int u0[4];
#pragma unroll
    for (int i = 0; i < 4; ++i) u0[i] = (wave + 8 * i) * 16;

    v8f c[RT_][4] = {};              // f32 accumulators: 4 row-tiles x 4 u-tiles

    const float2* As2 = (const float2*)As;
    for (int k = 0; k < D_; k += 4) {
        const int kb = k + 2 * half;
        // A 16x4 fragments for the 4 row-tiles (one b64 LDS load each)
        v2f a[RT_];
#pragma unroll
        for (int rt = 0; rt < RT_; ++rt) {
            const float2 af = As2[((rt * 16 + l16) * D_ + kb) >> 1];
            a[rt].x = af.x; a[rt].y = af.y;
        }
        // B 4x16 fragments for the 4 u-tiles; reused across all row-tiles
#pragma unroll
        for (int i = 0; i < 4; ++i) {
            const int col = u0[i] + l16;
            v2f bf;
            bf.x = W1[(size_t)kb * U_ + col];
            bf.y = W1[(size_t)(kb + 1) * U_ + col];
#pragma unroll
            for (int rt = 0; rt < RT_; ++rt)
                c[rt][i] = __builtin_amdgcn_wmma_f32_16x16x4_f32(
                    false, a[rt], false, bf, (short)0, c[rt][i], false, false);
        }
    }

    // Epilogue: acc[rt][r] = partial score for row rt*16 + (r + 8*half),
    // summed over this wave's 4 u-tiles (lane's u = u0[i]+l16 fixed per C-reg).
    float acc[RT_][8];
#pragma unroll
    for (int rt = 0; rt < RT_; ++rt)
#pragma unroll
        for (int r = 0; r < 8; ++r) acc[rt][r] = 0.0f;

#pragma unroll
    for (int i = 0; i < 4; ++i) {
        const int u = u0[i] + l16;
        const float qv = qbuf[b * U_ + u];
        const float vv = V[u];
#pragma unroll
        for (int rt = 0; rt < RT_; ++rt)
#pragma unroll
            for (int r = 0; r < 8; ++r)
                acc[rt][r] += tanhf(c[rt][i][r] + qv) * vv;
    }

    // Reduce over the 16 N-lanes within each half-wave (xor masks stay in-half).
#pragma unroll
    for (int rt = 0; rt < RT_; ++rt)
#pragma unroll
        for (int r = 0; r < 8; ++r) {
            float v = acc[rt][r];
            v += __shfl_xor(v, 1, 32);
            v += __shfl_xor(v, 2, 32);
            v += __shfl_xor(v, 4, 32);
            v += __shfl_xor(v, 8, 32);
            acc[rt][r] = v;
        }
    if (l16 == 0) {
#pragma unroll
        for (int rt = 0; rt < RT_; ++rt)
#pragma unroll
            for (int r = 0; r < 8; ++r)
                atomicAdd(&sc[rt * 16 + r + 8 * half], acc[rt][r]);  // ds_add_f32
    }
    __syncthreads();
    if (tid < ROWS_) scores[rowBase + tid] = sc[tid] + bv[0];
}

// ---------------------------------------------------------------------------
// K2: softmax over S per batch. One workgroup (256 threads) per b.
// ---------------------------------------------------------------------------
__global__ __launch_bounds__(256) void softmax_kernel(
    const float* __restrict__ scores, float* __restrict__ attn)
{
    __shared__ float red[256];
    const int b = blockIdx.x;
    const int tid = threadIdx.x;
    const float* row = scores + (size_t)b * S_;
    float* out = attn + (size_t)b * S_;

    float m = -INFINITY;
#pragma unroll
    for (int j = 0; j < S_ / 256; ++j) m = fmaxf(m, row[tid + j * 256]);
    red[tid] = m; __syncthreads();
    for (int off = 128; off > 0; off >>= 1) {
        if (tid < off) red[tid] = fmaxf(red[tid], red[tid + off]);
        __syncthreads();
    }
    m = red[0]; __syncthreads();

    float s = 0.0f;
#pragma unroll
    for (int j = 0; j < S_ / 256; ++j) {
        const float e = expf(row[tid + j * 256] - m);
        out[tid + j * 256] = e;
        s += e;
    }
    red[tid] = s; __syncthreads();
    for (int off = 128; off > 0; off >>= 1) {
        if (tid < off) red[tid] += red[tid + off];
        __syncthreads();
    }
    const float inv = 1.0f / red[0];
#pragma unroll
    for (int j = 0; j < S_ / 256; ++j) out[tid + j * 256] *= inv;
}

// ---------------------------------------------------------------------------
// K3: context[b,d] = sum_s attn[b,s] * enc[b,s,d]   (HBM-streaming, coalesced)
// ---------------------------------------------------------------------------
__global__ __launch_bounds__(256) void context_kernel(
    const float* __restrict__ enc, const float* __restrict__ attn,
    float* __restrict__ ctx)
{
    const int b = blockIdx.x >> 1;
    const int d = ((blockIdx.x & 1) << 8) + threadIdx.x;
    const float* w = attn + (size_t)b * S_;
    const float* e = enc + ((size_t)b * S_) * D_ + d;
    float acc = 0.0f;
#pragma unroll 8
    for (int s = 0; s < S_; ++s)
        acc += w[s] * e[(size_t)s * D_];
    ctx[b * D_ + d] = acc;
}

// ---------------------------------------------------------------------------
extern "C" void kernel_launch(void* const* d_in, const int* in_sizes, int n_in,
                              void* d_out, int out_size, void* d_ws, size_t ws_size,
                              hipStream_t stream)
{
    (void)in_sizes; (void)n_in; (void)out_size; (void)ws_size;

    const float* enc = (const float*)d_in[0];   // [B,S,D]
    const float* dec = (const float*)d_in[1];   // [B,D]
    const float* W1  = (const float*)d_in[2];   // [D,U]
    const float* b1  = (const float*)d_in[3];   // [U]
    const float* W2  = (const float*)d_in[4];   // [D,U]
    const float* b2  = (const float*)d_in[5];   // [U]
    const float* V   = (const float*)d_in[6];   // [U,1]
    const float* bv  = (const float*)d_in[7];   // [1]

    float* out  = (float*)d_out;
    float* ctx  = out;                 // context_vector  [B,D]  (first output)
    float* attn = out + B_ * D_;       // attention_weights [B,S,1] (second)

    float* qbuf   = (float*)d_ws;                                   // [B,U]
    float* scores = (float*)((char*)d_ws + (size_t)B_ * U_ * 4);    // [B,S]

    dec_proj_kernel  <<<B_ * 2,              256, 0, stream>>>(dec, W2, b1, b2, qbuf);
    attn_score_kernel<<<(B_ * S_) / ROWS_,   256, 0, stream>>>(enc, W1, qbuf, V, bv, scores);
    softmax_kernel   <<<B_,                  256, 0, stream>>>(scores, attn);
    context_kernel   <<<B_ * 2,              256, 0, stream>>>(enc, attn, ctx);
}